// EdgewiseGNNLayer_88399016886797
// MI455X (gfx1250) — compile-verified
//
#include <hip/hip_runtime.h>
#include <hip/hip_bf16.h>

typedef __attribute__((ext_vector_type(2))) float v2f;
typedef __attribute__((ext_vector_type(8))) float v8f;

#define NNODES 100000
#define NEDGES 1600000
#define DFEAT  128
#define NTYPE  5
#define KDIM   (NTYPE * DFEAT)      /* 640 */
#define KSTEPS (KDIM / 4)           /* 160 */
#define EPSV   1e-5f

/* ---- workspace layout (float offsets) ---- */
#define OFF_AGG   ((size_t)0)
#define SZ_AGG    ((size_t)NNODES * NTYPE * DFEAT)   /* 64,000,000 */
#define OFF_DEG   (OFF_AGG + SZ_AGG)
#define SZ_DEG    ((size_t)NNODES * NTYPE)           /* 500,000 */
#define OFF_GSUM  (OFF_DEG + SZ_DEG)
#define OFF_GSQ   (OFF_GSUM + DFEAT)
#define OFF_SCALE (OFF_GSQ + DFEAT)
#define OFF_SHIFT (OFF_SCALE + DFEAT)
#define OFF_H     (OFF_SHIFT + DFEAT)
#define SZ_H      ((size_t)NNODES * DFEAT)           /* 12,800,000 */
#define OFF_H1    (OFF_H + SZ_H)
#define OFF_BPACK (OFF_H1 + SZ_H)
#define SZ_BPACK  ((size_t)KSTEPS * 8 * 32 * 2)      /* 81,920 floats = 320KB */
#define ZERO_SPAN (SZ_AGG + SZ_DEG + 2 * DFEAT)      /* agg+deg+gsum+gsq */

/* ---------------- zero (float4) ---------------- */
__global__ void zero4_kernel(float4* __restrict__ p, size_t n4) {
    size_t stride = (size_t)gridDim.x * blockDim.x;
    for (size_t i = (size_t)blockIdx.x * blockDim.x + threadIdx.x; i < n4; i += stride)
        p[i] = make_float4(0.f, 0.f, 0.f, 0.f);
}

/* ---------------- per-column sum / sumsq ---------------- */
__global__ void colstats_kernel(const float* __restrict__ x,
                                float* __restrict__ gsum, float* __restrict__ gsq, int n) {
    __shared__ float ls[256];
    __shared__ float ls2[256];
    int col  = threadIdx.x & 127;
    int half = threadIdx.x >> 7;
    float s = 0.f, s2 = 0.f;
    for (int r = blockIdx.x * 2 + half; r < n; r += gridDim.x * 2) {
        float v = x[(size_t)r * DFEAT + col];
        s += v; s2 += v * v;
    }
    ls[threadIdx.x] = s; ls2[threadIdx.x] = s2;
    __syncthreads();
    if (threadIdx.x < 128) {
        atomicAdd(gsum + col, ls[threadIdx.x] + ls[threadIdx.x + 128]);
        atomicAdd(gsq  + col, ls2[threadIdx.x] + ls2[threadIdx.x + 128]);
    }
}

/* ---------------- BN scale/shift finalize ---------------- */
__global__ void finalize_kernel(const float* __restrict__ gsum, const float* __restrict__ gsq,
                                const float* __restrict__ g, const float* __restrict__ be,
                                float* __restrict__ scale, float* __restrict__ shift, int n) {
    int j = threadIdx.x;
    if (j < DFEAT) {
        float inv = 1.0f / (float)n;
        float mu  = gsum[j] * inv;
        float var = gsq[j] * inv - mu * mu;
        float sc  = g[j] * rsqrtf(var + EPSV);
        scale[j] = sc;
        shift[j] = be[j] - mu * sc;
    }
}

/* ---------------- h = relu(scale*x + shift), float4 ---------------- */
__global__ void bnrelu_kernel(const float4* __restrict__ x, const float4* __restrict__ sc4,
                              const float4* __restrict__ sh4, float4* __restrict__ h, size_t n4) {
    size_t stride = (size_t)gridDim.x * blockDim.x;
    for (size_t i = (size_t)blockIdx.x * blockDim.x + threadIdx.x; i < n4; i += stride) {
        int c = (int)(i & 31);           /* 32 float4 per 128-wide row */
        float4 v = x[i], s = sc4[c], b = sh4[c];
        float4 r;
        r.x = fmaxf(0.f, fmaf(v.x, s.x, b.x));
        r.y = fmaxf(0.f, fmaf(v.y, s.y, b.y));
        r.z = fmaxf(0.f, fmaf(v.z, s.z, b.z));
        r.w = fmaxf(0.f, fmaf(v.w, s.w, b.w));
        h[i] = r;
    }
}

/* ---------------- pack weights into WMMA-B fragment order ----------------
 * bpack float2[idx], idx = (kstep*8 + jt)*32 + lane
 * lane<16 : (B[k0+0][j], B[k0+1][j]) ; lane>=16 : (B[k0+2][j], B[k0+3][j])
 * with j = jt*16 + (lane&15), B[k=t*128+d][j] = w[(t*128+j)*128 + d]      */
__global__ void packb_kernel(const float* __restrict__ w, float2* __restrict__ bpack) {
    int idx = blockIdx.x * blockDim.x + threadIdx.x;
    if (idx >= KSTEPS * 8 * 32) return;
    int lane  = idx & 31;
    int jt    = (idx >> 5) & 7;
    int kstep = idx >> 8;
    int k0 = kstep * 4 + ((lane >> 4) << 1);
    int j  = jt * 16 + (lane & 15);
    int t0 = k0 >> 7, d0 = k0 & 127;
    int t1 = (k0 + 1) >> 7, d1 = (k0 + 1) & 127;
    float2 o;
    o.x = w[((size_t)(t0 * DFEAT + j)) * DFEAT + d0];
    o.y = w[((size_t)(t1 * DFEAT + j)) * DFEAT + d1];
    bpack[idx] = o;
}

/* ---------------- edge scatter: agg[dst,etype] += h[src]; deg count ----- */
__global__ void edge_scatter_kernel(const float4* __restrict__ h4,
                                    const int* __restrict__ src, const int* __restrict__ dst,
                                    const int* __restrict__ et,
                                    float* __restrict__ agg, float* __restrict__ deg, int ne) {
    int lane = threadIdx.x & 31;
    int gw   = (blockIdx.x * blockDim.x + threadIdx.x) >> 5;
    int nw   = (gridDim.x * blockDim.x) >> 5;
    for (int e = gw; e < ne; e += nw) {
        int s = src[e], d = dst[e], t = et[e];
        float4 v = h4[(size_t)s * 32 + lane];
        float* p = agg + (((size_t)d * NTYPE + t) << 7) + lane * 4;
        atomicAdd(p + 0, v.x);
        atomicAdd(p + 1, v.y);
        atomicAdd(p + 2, v.z);
        atomicAdd(p + 3, v.w);
        if (lane == 0) atomicAdd(deg + (size_t)d * NTYPE + t, 1.0f);
    }
}

/* ---------------- WMMA GEMM: out = agg[N,640] x B[640,128] (+deg*b)(+resid) */
__global__ __launch_bounds__(256)
void gemm_wmma_kernel(const float* __restrict__ agg, const float2* __restrict__ bpack,
                      const float* __restrict__ deg, const float* __restrict__ bias,
                      const float* __restrict__ resid, float* __restrict__ out, int nrows) {
    __shared__ float2 lds[32 * 8 * 32];           /* 64 KB: one 32-kstep chunk of B */
    const int wave = threadIdx.x >> 5;
    const int lane = threadIdx.x & 31;
    const int rowbase = blockIdx.x * 128 + wave * 16;
    const bool active = (rowbase < nrows);        /* wave-uniform; nrows % 16 == 0 */

    v8f acc[8];
#pragma unroll
    for (int jt = 0; jt < 8; ++jt) acc[jt] = 0.0f;

    const int mrow  = lane & 15;
    const int khalf = (lane >> 4) << 1;           /* 0 or 2 */
    const float* arow = agg + (size_t)(rowbase + mrow) * KDIM + khalf;

    for (int chunk = 0; chunk < 5; ++chunk) {
        __syncthreads();
        const float4* s4 = (const float4*)(bpack + (size_t)chunk * (32 * 8 * 32));
        float4* d4 = (float4*)lds;
        for (int i = threadIdx.x; i < 4096; i += 256) d4[i] = s4[i];
        __syncthreads();
        if (active) {
            for (int ks = 0; ks < 32; ++ks) {
                const int k0 = (chunk * 32 + ks) * 4;
                v2f a = *(const v2f*)(arow + k0);
                const float2* bl = lds + ks * 256 + lane;
#pragma unroll
                for (int jt = 0; jt < 8; ++jt) {
                    float2 bf = bl[jt * 32];
                    v2f b; b.x = bf.x; b.y = bf.y;
                    acc[jt] = __builtin_amdgcn_wmma_f32_16x16x4_f32(
                        false, a, false, b, (short)0, acc[jt], false, false);
                }
            }
        }
    }
    if (!active) return;

    const int colbase = lane & 15;
    const int rowoff  = (lane >> 4) * 8;
#pragma unroll
    for (int v = 0; v < 8; ++v) {
        int row = rowbase + rowoff + v;
        float d0 = deg[(size_t)row * NTYPE + 0];
        float d1 = deg[(size_t)row * NTYPE + 1];
        float d2 = deg[(size_t)row * NTYPE + 2];
        float d3 = deg[(size_t)row * NTYPE + 3];
        float d4 = deg[(size_t)row * NTYPE + 4];
#pragma unroll
        for (int jt = 0; jt < 8; ++jt) {
            int j = jt * 16 + colbase;
            float val = acc[jt][v];
            val += d0 * bias[0 * DFEAT + j] + d1 * bias[1 * DFEAT + j] +
                   d2 * bias[2 * DFEAT + j] + d3 * bias[3 * DFEAT + j] +
                   d4 * bias[4 * DFEAT + j];
            if (resid) val += resid[(size_t)row * DFEAT + j];
            out[(size_t)row * DFEAT + j] = val;
        }
    }
}

/* ---------------- host-side round driver ---------------- */
static void run_round(const float* x, const float* w, const float* b,
                      const float* g, const float* be, const float* resid,
                      float* outp, float* ws, const int* src, const int* dst,
                      const int* etype, hipStream_t stream) {
    zero4_kernel<<<4096, 256, 0, stream>>>((float4*)(ws + OFF_AGG), ZERO_SPAN / 4);
    colstats_kernel<<<1024, 256, 0, stream>>>(x, ws + OFF_GSUM, ws + OFF_GSQ, NNODES);
    finalize_kernel<<<1, 128, 0, stream>>>(ws + OFF_GSUM, ws + OFF_GSQ, g, be,
                                           ws + OFF_SCALE, ws + OFF_SHIFT, NNODES);
    bnrelu_kernel<<<2048, 256, 0, stream>>>((const float4*)x, (const float4*)(ws + OFF_SCALE),
                                            (const float4*)(ws + OFF_SHIFT),
                                            (float4*)(ws + OFF_H), SZ_H / 4);
    packb_kernel<<<(KSTEPS * 8 * 32 + 255) / 256, 256, 0, stream>>>(w, (float2*)(ws + OFF_BPACK));
    edge_scatter_kernel<<<4096, 256, 0, stream>>>((const float4*)(ws + OFF_H), src, dst, etype,
                                                  ws + OFF_AGG, ws + OFF_DEG, NEDGES);
    gemm_wmma_kernel<<<(NNODES + 127) / 128, 256, 0, stream>>>(
        ws + OFF_AGG, (const float2*)(ws + OFF_BPACK), ws + OFF_DEG, b, resid, outp, NNODES);
}

extern "C" void kernel_launch(void* const* d_in, const int* in_sizes, int n_in,
                              void* d_out, int out_size, void* d_ws, size_t ws_size,
                              hipStream_t stream) {
    const float* features = (const float*)d_in[0];
    const int*   src      = (const int*)d_in[1];
    const int*   dst      = (const int*)d_in[2];
    const int*   etype    = (const int*)d_in[3];
    const float* w1 = (const float*)d_in[4];
    const float* b1 = (const float*)d_in[5];
    const float* g1 = (const float*)d_in[6];
    const float* be1 = (const float*)d_in[7];
    const float* w2 = (const float*)d_in[8];
    const float* b2 = (const float*)d_in[9];
    const float* g2 = (const float*)d_in[10];
    const float* be2 = (const float*)d_in[11];
    float* ws  = (float*)d_ws;
    float* out = (float*)d_out;

    /* round 1: features -> h1 (stored in workspace) */
    run_round(features, w1, b1, g1, be1, /*resid=*/nullptr,
              ws + OFF_H1, ws, src, dst, etype, stream);
    /* round 2: h1 -> d_out, residual += features */
    run_round(ws + OFF_H1, w2, b2, g2, be2, /*resid=*/features,
              out, ws, src, dst, etype, stream);
    (void)in_sizes; (void)n_in; (void)out_size; (void)ws_size;
}